// SSDLoss_72378788873057
// MI455X (gfx1250) — compile-verified
//
#include <hip/hip_runtime.h>
#include <stdint.h>

#define NB 64
#define NA 24564
#define NO 32
#define NC 21
#define FTHRESH 0.5f
#define VAR0 0.1f
#define VAR1 0.2f
#define NEGPOS 3

typedef float v2f __attribute__((ext_vector_type(2)));
typedef float v8f __attribute__((ext_vector_type(8)));

// ---------------------------------------------------------------------------
// Exact wave32 sum via V_WMMA_F32_16X16X4_F32 (ones-matmul reduction).
// Stage 1: A[m][0]=v_m (lanes 0-15), A[m][2]=v_{m+16} (lanes 16-31), A[.][1,3]=0,
//          B=ones -> D[m][n] = v_m + v_{m+16} = u_m  (replicated over n).
// D vgpr r holds u_r (lanes 0-15) / u_{r+8} (lanes 16-31); feeding vgpr pairs
// (d0,d1),(d2,d3),(d4,d5),(d6,d7) back as A with B=ones and accumulating C
// sums all u_m -> total replicated in every lane of acc.
// EXEC must be all ones at the call site (callers guarantee convergence).
// ---------------------------------------------------------------------------
__device__ __forceinline__ float wave_sum32(float v) {
  v2f a;   a[0] = v;    a[1] = 0.0f;
  v2f one; one[0] = 1.0f; one[1] = 1.0f;
  v8f z = {0.f,0.f,0.f,0.f,0.f,0.f,0.f,0.f};
  v8f u = __builtin_amdgcn_wmma_f32_16x16x4_f32(false, a, false, one, (short)0, z, false, false);
  v8f acc = {0.f,0.f,0.f,0.f,0.f,0.f,0.f,0.f};
  v2f p;
  p[0] = u[0]; p[1] = u[1];
  acc = __builtin_amdgcn_wmma_f32_16x16x4_f32(false, p, false, one, (short)0, acc, false, false);
  p[0] = u[2]; p[1] = u[3];
  acc = __builtin_amdgcn_wmma_f32_16x16x4_f32(false, p, false, one, (short)0, acc, false, false);
  p[0] = u[4]; p[1] = u[5];
  acc = __builtin_amdgcn_wmma_f32_16x16x4_f32(false, p, false, one, (short)0, acc, false, false);
  p[0] = u[6]; p[1] = u[7];
  acc = __builtin_amdgcn_wmma_f32_16x16x4_f32(false, p, false, one, (short)0, acc, false, false);
  return acc[0];
}

__device__ __forceinline__ float sl1(float d) {
  float a = fabsf(d);
  return a < 1.0f ? 0.5f * a * a : a - 0.5f;
}

// ---------------------------------------------------------------------------
__global__ void k_init(unsigned long long* bestkey, float* nposf, float* lossl, float* lossc) {
  int i = blockIdx.x * blockDim.x + threadIdx.x;
  if (i < NB * NO) bestkey[i] = 0ull;
  if (i < NB) nposf[i] = 0.0f;
  if (i == 0) { lossl[0] = 0.0f; lossc[0] = 0.0f; }
}

// ---------------------------------------------------------------------------
// Per (b, anchor): IoU vs all 32 truths; per-anchor max/argmax (first-index
// ties), per-truth global argmax via packed u64 atomicMax with ~a tiebreak.
// ---------------------------------------------------------------------------
__global__ void __launch_bounds__(256)
k_match(const float* __restrict__ anchors, const float* __restrict__ targets,
        float* __restrict__ bto, int* __restrict__ bti, int* __restrict__ hit,
        unsigned long long* __restrict__ bestkey) {
  __shared__ float tb[NO * 5];
  __shared__ unsigned long long wkey[NO];
  const int b = blockIdx.y;
  const int a = blockIdx.x * 256 + threadIdx.x;
  for (int i = threadIdx.x; i < NO * 5; i += 256) tb[i] = targets[(size_t)b * NO * 5 + i];
  if (threadIdx.x < NO) wkey[threadIdx.x] = 0ull;
  __syncthreads();

  const bool valid = a < NA;
  const int ai = valid ? a : (NA - 1);
  float4 an = ((const float4*)anchors)[ai];
  float px1 = an.x - an.z * 0.5f, py1 = an.y - an.w * 0.5f;
  float px2 = an.x + an.z * 0.5f, py2 = an.y + an.w * 0.5f;
  float areaB = (px2 - px1) * (py2 - py1);

  float bestv = -1.0f; int besti = 0;
#pragma unroll 4
  for (int t = 0; t < NO; ++t) {
    float x1 = tb[t * 5 + 0], y1 = tb[t * 5 + 1], x2 = tb[t * 5 + 2], y2 = tb[t * 5 + 3];
    float iw = fmaxf(fminf(px2, x2) - fmaxf(px1, x1), 0.0f);
    float ih = fmaxf(fminf(py2, y2) - fmaxf(py1, y1), 0.0f);
    float inter = iw * ih;
    float areaA = (x2 - x1) * (y2 - y1);
    float iou = inter / (areaA + areaB - inter);
    if (iou > bestv) { bestv = iou; besti = t; }
    if (valid) {
      unsigned long long key =
          ((unsigned long long)__float_as_uint(iou) << 32) | (unsigned)(~(unsigned)a);
      atomicMax(&wkey[t], key);
    }
  }
  if (valid) {
    bto[(size_t)b * NA + a] = bestv;
    bti[(size_t)b * NA + a] = besti;
    hit[(size_t)b * NA + a] = 0;
  }
  __syncthreads();
  if (threadIdx.x < NO) atomicMax(&bestkey[b * NO + threadIdx.x], wkey[threadIdx.x]);
}

// ---------------------------------------------------------------------------
__global__ void k_claim(const unsigned long long* __restrict__ bestkey,
                        float* __restrict__ bto, int* __restrict__ hit) {
  int i = blockIdx.x * blockDim.x + threadIdx.x;
  if (i >= NB * NO) return;
  int b = i / NO, t = i % NO;
  unsigned long long key = bestkey[i];
  unsigned astar = ~(unsigned)(key & 0xffffffffull);
  bto[(size_t)b * NA + astar] = 2.0f;                 // .at[best_anchor_idx].set(2.0)
  atomicMax(&hit[(size_t)b * NA + astar], t + 1);     // .at[].max(arange(1,n+1))
}

// ---------------------------------------------------------------------------
// Per-anchor losses. conf tile (<=256 anchors x 21 f32) staged into LDS with
// global_load_async_to_lds_b128 (base rounded down to 16B; total conf bytes
// = 64*24564*84 is a multiple of 16, so the rounded-up chunk range never
// leaves the buffer). LDS row reads are bank-conflict-free (stride 21 dwords,
// gcd(21,32)=1). Wave-level reductions of loss_l / pos-count via WMMA.
// ---------------------------------------------------------------------------
__global__ void __launch_bounds__(256)
k_loss(const float* __restrict__ loc, const float* __restrict__ conf,
       const float* __restrict__ anchors, const float* __restrict__ targets,
       const float* __restrict__ bto, const int* __restrict__ bti, const int* __restrict__ hit,
       float* __restrict__ score, float* __restrict__ ce,
       float* __restrict__ nposf, float* __restrict__ lossl) {
  __shared__ __align__(16) unsigned char tile[1345 * 16];  // 21520 B
  const int b = blockIdx.y;
  const int a0 = blockIdx.x * 256;
  const int tid = threadIdx.x;
  const int a = a0 + tid;
  const int nanch = min(256, NA - a0);

  // ---- async stage conf tile into LDS ----
  unsigned long long gbyte0 = ((unsigned long long)b * NA + (unsigned)a0) * (NC * 4);
  unsigned long long s0 = gbyte0 & ~15ull;
  unsigned shift = (unsigned)(gbyte0 - s0);            // 0,4,8,12
  unsigned nbytes = (unsigned)nanch * (NC * 4) + shift;
  unsigned nchunks = (nbytes + 15u) >> 4;
  unsigned long long gbase = (unsigned long long)(uintptr_t)conf + s0;
  unsigned ldsbase = (unsigned)(uintptr_t)&tile[0];
  for (unsigned c = tid; c < nchunks; c += 256u) {
    unsigned lds = ldsbase + (c << 4);
    unsigned voff = c << 4;
    asm volatile("global_load_async_to_lds_b128 %0, %1, %2"
                 :: "v"(lds), "v"(voff), "s"(gbase) : "memory");
  }
  asm volatile("s_wait_asynccnt 0" ::: "memory");
  __syncthreads();

  // ---- match outputs -> conf_t / loc_t ----
  const bool valid = tid < nanch;
  const int ai = valid ? a : (NA - 1);
  float ov = bto[(size_t)b * NA + ai];
  int h = hit[(size_t)b * NA + ai];
  int idx = (h > 0) ? (h - 1) : bti[(size_t)b * NA + ai];
  const float* tgt = targets + ((size_t)b * NO + idx) * 5;
  float x1 = tgt[0], y1 = tgt[1], x2 = tgt[2], y2 = tgt[3], lab = tgt[4];
  int conft = (ov < FTHRESH) ? 0 : ((int)lab + 1);
  bool pos = conft > 0;

  float4 an = ((const float4*)anchors)[ai];
  float gcx = ((x1 + x2) * 0.5f - an.x) / (an.z * VAR0);
  float gcy = ((y1 + y2) * 0.5f - an.y) / (an.w * VAR0);
  float gw  = logf((x2 - x1) / an.z + 1e-5f) / VAR1;
  float gh  = logf((y2 - y1) / an.w + 1e-5f) / VAR1;
  float4 l = ((const float4*)loc)[(size_t)b * NA + ai];
  float ll = sl1(l.x - gcx) + sl1(l.y - gcy) + sl1(l.z - gw) + sl1(l.w - gh);
  float llc = (valid && pos) ? ll : 0.0f;
  float npc = (valid && pos) ? 1.0f : 0.0f;

  // ---- log-softmax over the 21 classes from LDS ----
  const float* cr = (const float*)(tile + shift + (unsigned)tid * (NC * 4));
  float m = cr[0];
#pragma unroll
  for (int c = 1; c < NC; ++c) m = fmaxf(m, cr[c]);
  float s = 0.0f;
#pragma unroll
  for (int c = 0; c < NC; ++c) s += expf(cr[c] - m);
  float lse = m + logf(s);
  float cei = lse - cr[conft];                 // -logp[conf_t]
  float sc  = pos ? -1.0f : (lse - cr[0]);     // sentinel<0 marks positives
  if (valid) {
    score[(size_t)b * NA + a] = sc;
    ce[(size_t)b * NA + a] = cei;
  }

  // ---- WMMA wave reductions (all lanes converged here) ----
  float wl = wave_sum32(llc);
  float wn = wave_sum32(npc);
  if ((tid & 31) == 0) {
    atomicAdd(lossl, wl);
    atomicAdd(&nposf[b], wn);
  }
}

// ---------------------------------------------------------------------------
// Exact top-k selection per batch: 3-pass radix select (11/11/10 bits) over
// key = (score<0 ? 0 : bits(score)), then sum ce over positives + selected
// negatives; equal-key ties resolved by ascending index (stable argsort).
// ---------------------------------------------------------------------------
__global__ void __launch_bounds__(1024)
k_select(const float* __restrict__ score, const float* __restrict__ ce,
         const float* __restrict__ nposf, float* __restrict__ lossc) {
  __shared__ unsigned hist[2048];
  __shared__ unsigned sBin, sBefore;
  __shared__ unsigned eqList[256];
  __shared__ unsigned eqCount;
  __shared__ float wsum[32];
  const int b = blockIdx.x;
  const int tid = threadIdx.x;
  const float* sc = score + (size_t)b * NA;
  const float* cearr = ce + (size_t)b * NA;
  int np = (int)(nposf[b] + 0.5f);
  unsigned k = (unsigned)min(NEGPOS * np, NA - 1);

  // pass 1: top 11 bits
  for (int i = tid; i < 2048; i += 1024) hist[i] = 0u;
  if (tid == 0) eqCount = 0u;
  __syncthreads();
  for (int i = tid; i < NA; i += 1024) {
    float v = sc[i];
    unsigned key = (v < 0.0f) ? 0u : __float_as_uint(v);
    atomicAdd(&hist[key >> 21], 1u);
  }
  __syncthreads();
  if (tid == 0) {
    unsigned acc = 0, before = 0; int bsel = 0;
    for (int i = 2047; i >= 0; --i) {
      if (acc + hist[i] >= k) { bsel = i; before = acc; break; }
      acc += hist[i];
    }
    sBin = (unsigned)bsel; sBefore = before;
  }
  __syncthreads();
  unsigned b1 = sBin, k1 = k - sBefore;

  // pass 2: mid 11 bits within bin b1
  __syncthreads();
  for (int i = tid; i < 2048; i += 1024) hist[i] = 0u;
  __syncthreads();
  for (int i = tid; i < NA; i += 1024) {
    float v = sc[i];
    unsigned key = (v < 0.0f) ? 0u : __float_as_uint(v);
    if ((key >> 21) == b1) atomicAdd(&hist[(key >> 10) & 0x7FFu], 1u);
  }
  __syncthreads();
  if (tid == 0) {
    unsigned acc = 0, before = 0; int bsel = 0;
    for (int i = 2047; i >= 0; --i) {
      if (acc + hist[i] >= k1) { bsel = i; before = acc; break; }
      acc += hist[i];
    }
    sBin = (unsigned)bsel; sBefore = before;
  }
  __syncthreads();
  unsigned b2 = sBin, k2 = k1 - sBefore;

  // pass 3: low 10 bits within (b1,b2)
  __syncthreads();
  for (int i = tid; i < 1024; i += 1024) hist[i] = 0u;
  __syncthreads();
  unsigned hi21 = (b1 << 11) | b2;
  for (int i = tid; i < NA; i += 1024) {
    float v = sc[i];
    unsigned key = (v < 0.0f) ? 0u : __float_as_uint(v);
    if ((key >> 10) == hi21) atomicAdd(&hist[key & 0x3FFu], 1u);
  }
  __syncthreads();
  if (tid == 0) {
    unsigned acc = 0, before = 0; int bsel = 0;
    for (int i = 1023; i >= 0; --i) {
      if (acc + hist[i] >= k2) { bsel = i; before = acc; break; }
      acc += hist[i];
    }
    sBin = (unsigned)bsel; sBefore = before;
  }
  __syncthreads();
  unsigned b3 = sBin;
  unsigned needeq = k2 - sBefore;
  unsigned T = (b1 << 21) | (b2 << 10) | b3;

  // pass 4: sum ce over pos + (key > T); collect key==T ties
  float sum = 0.0f;
  for (int i = tid; i < NA; i += 1024) {
    float v = sc[i];
    float c = cearr[i];
    if (v < 0.0f) {
      sum += c;  // positive anchor
    } else {
      unsigned key = __float_as_uint(v);
      if (key > T) sum += c;
      else if (key == T) {
        unsigned p = atomicAdd(&eqCount, 1u);
        if (p < 256u) eqList[p] = (unsigned)i;
      }
    }
  }
  float w = wave_sum32(sum);  // all 1024 lanes converged
  if ((tid & 31) == 0) wsum[tid >> 5] = w;
  __syncthreads();
  if (tid == 0) {
    float tot = 0.0f;
    for (int j = 0; j < 32; ++j) tot += wsum[j];
    unsigned n = min(eqCount, 256u);
    unsigned take = min(needeq, n);
    for (unsigned r = 0; r < take; ++r) {   // needeq smallest indices among ties
      unsigned bestv = 0xFFFFFFFFu; int bj = -1;
      for (unsigned j = 0; j < n; ++j)
        if (eqList[j] < bestv) { bestv = eqList[j]; bj = (int)j; }
      tot += cearr[bestv];
      eqList[bj] = 0xFFFFFFFFu;
    }
    atomicAdd(lossc, tot);
  }
}

// ---------------------------------------------------------------------------
__global__ void k_final(const float* __restrict__ nposf, const float* __restrict__ lossl,
                        const float* __restrict__ lossc, float* __restrict__ out) {
  if (threadIdx.x == 0 && blockIdx.x == 0) {
    float tp = 0.0f;
    for (int i = 0; i < NB; ++i) tp += nposf[i];
    float denom = tp * (float)NC;
    out[0] = lossc[0] / denom;
    out[1] = lossl[0] / denom;
  }
}

// ---------------------------------------------------------------------------
extern "C" void kernel_launch(void* const* d_in, const int* in_sizes, int n_in,
                              void* d_out, int out_size, void* d_ws, size_t ws_size,
                              hipStream_t stream) {
  const float* loc     = (const float*)d_in[0];  // (B,A,4)
  const float* conf    = (const float*)d_in[1];  // (B,A,21)
  const float* anchors = (const float*)d_in[2];  // (A,4)
  const float* targets = (const float*)d_in[3];  // (B,O,5)
  float* out = (float*)d_out;

  char* w = (char*)d_ws;
  size_t off = 0;
  auto carve = [&](size_t bytes) -> void* {
    void* p = (void*)(w + off);
    off += (bytes + 255) & ~(size_t)255;
    return p;
  };
  float* bto   = (float*)carve(sizeof(float) * NB * NA);
  int*   bti   = (int*)  carve(sizeof(int)   * NB * NA);
  int*   hit   = (int*)  carve(sizeof(int)   * NB * NA);
  float* score = (float*)carve(sizeof(float) * NB * NA);
  float* ce    = (float*)carve(sizeof(float) * NB * NA);
  unsigned long long* bestkey = (unsigned long long*)carve(8 * NB * NO);
  float* nposf = (float*)carve(sizeof(float) * NB);
  float* lossl = (float*)carve(sizeof(float));
  float* lossc = (float*)carve(sizeof(float));

  dim3 gA((NA + 255) / 256, NB);
  k_init<<<(NB * NO + 255) / 256, 256, 0, stream>>>(bestkey, nposf, lossl, lossc);
  k_match<<<gA, 256, 0, stream>>>(anchors, targets, bto, bti, hit, bestkey);
  k_claim<<<(NB * NO + 255) / 256, 256, 0, stream>>>(bestkey, bto, hit);
  k_loss<<<gA, 256, 0, stream>>>(loc, conf, anchors, targets, bto, bti, hit,
                                 score, ce, nposf, lossl);
  k_select<<<NB, 1024, 0, stream>>>(score, ce, nposf, lossc);
  k_final<<<1, 64, 0, stream>>>(nposf, lossl, lossc, out);
}